// GraphSAGELayerImp1_48928267436078
// MI455X (gfx1250) — compile-verified
//
#include <hip/hip_runtime.h>

// GraphSAGE layer for MI455X (gfx1250, wave32).
// out = relu([feat[self] , mean_k feat[neigh]] @ W.T + b)
//
// Gather-bound (feature table 102 MB < 192 MB L2). f32 WMMA 16x16x4 keeps
// reference precision; compute is negligible. This version:
//  - packs W into per-lane WMMA-B float2 layout in d_ws (prep kernel)
//  - persistent workgroups stage packed B in LDS ONCE, grid-stride node tiles
//  - float4 (b128) gather loads, one A ds_load_b64 feeds two WMMAs

typedef float v2f __attribute__((ext_vector_type(2)));
typedef float v4f __attribute__((ext_vector_type(4)));
typedef float v8f __attribute__((ext_vector_type(8)));

#define F_IN   128
#define F2     256               // concat width (K of the GEMM)
#define F_OUT  128
#define KNB    16                // sampled neighbors
#define TILE_M 16                // nodes per tile
#define LDS_STRIDE 260           // 256+4 pad: stride mod 64 banks = 4 -> conflict-free A reads
#define NK4    (F2 / 4)          // 64 k-steps of 4
#define B_LDS_FLOATS (NK4 * 8 * 32 * 2)   // 32768 floats = 131072 B packed B
#define GRID_MAIN 1536

// ---- Kernel 1: pack W [F_OUT, 2F] into per-lane WMMA B operands.
// Bpack[(k0/4)*8 + c][lane] = float2{ W[c*16+n][ka], W[c*16+n][ka+1] }
//   n = lane&15, ka = k0 + ((lane>>4)<<1)   (ISA 7.12.2 32-bit B layout)
__global__ __launch_bounds__(256)
void sage_pack_B(const float* __restrict__ W, v2f* __restrict__ Bpack) {
    const int idx  = blockIdx.x * 256 + threadIdx.x;   // 0 .. 16383
    const int lane = idx & 31;
    const int c    = (idx >> 5) & 7;                   // output column tile
    const int k0   = (idx >> 8) << 2;                  // 0,4,...,252
    const int n    = lane & 15;
    const int ka   = k0 + ((lane >> 4) << 1);
    const float* wrow = W + (c * 16 + n) * F2;
    v2f v;
    v.x = wrow[ka];
    v.y = wrow[ka + 1];
    Bpack[idx] = v;
}

// ---- Kernel 2: persistent fused gather + mean + concat + WMMA GEMM + bias + relu
__global__ __launch_bounds__(128)
void sage_fused(const float* __restrict__ features,
                const int*   __restrict__ self_idx,
                const int*   __restrict__ neighbors,
                const v2f*   __restrict__ Bpack,
                const float* __restrict__ bias,
                float*       __restrict__ out,
                int ntiles) {
    extern __shared__ float smem[];
    float* Bl = smem;                    // packed B: 32768 floats (131 KB)
    float* h  = smem + B_LDS_FLOATS;     // concat tile: 16 x 260 floats

    const int t    = threadIdx.x;        // 0..127
    const int lane = t & 31;
    const int wave = t >> 5;             // 0..3

    // ---- Stage packed B to LDS once per (persistent) workgroup, coalesced b128
    {
        const v4f* src = (const v4f*)Bpack;
        v4f*       dst = (v4f*)Bl;
        for (int i = t; i < B_LDS_FLOATS / 4; i += 128) dst[i] = src[i];
    }

    // Per-thread constants
    const int r4  = wave;                // row group for gather phase (0..3)
    const int c4  = (t & 31) * 4;        // column (float4) for gather phase
    const int mn  = lane & 15;           // A-row / B-col inside 16x16 tile
    const int koff = (lane >> 4) << 1;   // 0 | 2 (K sub-offset per lane half)
    const int c0  = wave * 2;            // this wave's two output column tiles
    const int c1  = wave * 2 + 1;
    const float bv0 = bias[c0 * 16 + mn];
    const float bv1 = bias[c1 * 16 + mn];
    const v2f* Bw = (const v2f*)Bl;
    const float* hrow = h + mn * LDS_STRIDE;

    __syncthreads();

    for (int tile = blockIdx.x; tile < ntiles; tile += GRID_MAIN) {
        const int base = tile * TILE_M;

        // ---------- Phase 1: build h[16][256] (self | neighbor-mean), b128 loads
#pragma unroll
        for (int m0 = 0; m0 < TILE_M; m0 += 4) {
            const int m = m0 + r4;
            const int node = self_idx[base + m];
            *(v4f*)(h + m * LDS_STRIDE + c4) =
                *(const v4f*)(features + node * F_IN + c4);

            const int* nb = neighbors + (base + m) * KNB;
            v4f acc = {0.f, 0.f, 0.f, 0.f};
#pragma unroll
            for (int j = 0; j < KNB; ++j) {
                acc += *(const v4f*)(features + nb[j] * F_IN + c4);
            }
            acc *= (1.0f / (float)KNB);
            *(v4f*)(h + m * LDS_STRIDE + F_IN + c4) = acc;
        }
        __syncthreads();

        // ---------- Phase 2: two 16x16 output tiles per wave, K=256
        v8f acc0 = {bv0, bv0, bv0, bv0, bv0, bv0, bv0, bv0};
        v8f acc1 = {bv1, bv1, bv1, bv1, bv1, bv1, bv1, bv1};
#pragma unroll 4
        for (int k0 = 0; k0 < F2; k0 += 4) {
            const v2f a = *(const v2f*)(hrow + k0 + koff);      // one A feeds 2 WMMAs
            const v2f b0 = Bw[(((k0 >> 2) << 3) + c0) * 32 + lane];
            const v2f b1 = Bw[(((k0 >> 2) << 3) + c1) * 32 + lane];
            acc0 = __builtin_amdgcn_wmma_f32_16x16x4_f32(
                false, a, false, b0, (short)0, acc0, false, false);
            acc1 = __builtin_amdgcn_wmma_f32_16x16x4_f32(
                false, a, false, b1, (short)0, acc1, false, false);
        }

        // ---------- ReLU + store. C/D: VGPR r -> M=r | r+8 by lane half
        const int rowbase = base + ((lane >> 4) << 3);
#pragma unroll
        for (int r = 0; r < 8; ++r) {
            float v0 = acc0[r]; v0 = v0 > 0.f ? v0 : 0.f;
            float v1 = acc1[r]; v1 = v1 > 0.f ? v1 : 0.f;
            out[(rowbase + r) * F_OUT + c0 * 16 + mn] = v0;
            out[(rowbase + r) * F_OUT + c1 * 16 + mn] = v1;
        }
        __syncthreads();   // protect h before next tile overwrites it
    }
}

extern "C" void kernel_launch(void* const* d_in, const int* in_sizes, int n_in,
                              void* d_out, int out_size, void* d_ws, size_t ws_size,
                              hipStream_t stream) {
    const float* features  = (const float*)d_in[0];   // [200000,128] f32
    const int*   self_idx  = (const int*)  d_in[1];   // [100000]
    const int*   neighbors = (const int*)  d_in[2];   // [100000,16]
    const float* W         = (const float*)d_in[3];   // [128,256] f32
    const float* bias      = (const float*)d_in[4];   // [128] f32
    float*       out       = (float*)d_out;           // [100000,128] f32
    v2f*         Bpack     = (v2f*)d_ws;              // 16384 float2 = 128 KiB

    const int n_nodes = in_sizes[1];
    const int ntiles  = (n_nodes + TILE_M - 1) / TILE_M;   // 6250

    sage_pack_B<<<(NK4 * 8 * 32) / 256, 256, 0, stream>>>(W, Bpack);

    const size_t lds_bytes = (size_t)(B_LDS_FLOATS + TILE_M * LDS_STRIDE) * 4; // 147712 B
    const int grid = GRID_MAIN < ntiles ? GRID_MAIN : ntiles;
    sage_fused<<<grid, 128, lds_bytes, stream>>>(
        features, self_idx, neighbors, Bpack, bias, out, ntiles);
}